// MLPMixer_14310831030527
// MI455X (gfx1250) — compile-verified
//
#include <hip/hip_runtime.h>
#include <math.h>

#define PD      448     // padded spatial dim (7 * 64)
#define KPAD    448     // padded K for DIM-contraction (14 * 32)
#define NPADW   512     // padded N for DIM-sized weight matrices (4 * 128)
#define DIMV    427
#define TOKC    2048
#define BATCH   64
#define DEPTH_N 8

typedef float v2f __attribute__((ext_vector_type(2)));
typedef float v8f __attribute__((ext_vector_type(8)));

__device__ __forceinline__ float gelu_exact(float x) {
    return 0.5f * x * (1.0f + erff(x * 0.70710678118654752f));
}

// ---------------------------------------------------------------------------
// h[b,r,c] = mean over channel of x[b,:,r,c], zero-padded to [B,448,448]
// ---------------------------------------------------------------------------
__global__ void mean_c_kernel(const float* __restrict__ x, float* __restrict__ h) {
    long long idx = (long long)blockIdx.x * blockDim.x + threadIdx.x;
    const long long total = (long long)BATCH * PD * PD;
    if (idx >= total) return;
    int b  = (int)(idx / ((long long)PD * PD));
    int rc = (int)(idx % ((long long)PD * PD));
    int r = rc / PD, c = rc % PD;
    float v = 0.f;
    if (r < DIMV && c < DIMV) {
        long long base = ((long long)b * 3) * DIMV * DIMV + (long long)r * DIMV + c;
        v = (x[base] + x[base + (long long)DIMV * DIMV] + x[base + 2LL * DIMV * DIMV])
            * (1.0f / 3.0f);
    }
    h[idx] = v;
}

// ---------------------------------------------------------------------------
// Zero-padded weight copy: dst[d][r][c] = (r<sRows && c<sCols) ? src[d][r][c] : 0
// ---------------------------------------------------------------------------
__global__ void pad_w_kernel(const float* __restrict__ src, long long sStride,
                             int sRows, int sCols,
                             float* __restrict__ dst, long long dStride,
                             int dRows, int dCols, int depth) {
    long long idx = (long long)blockIdx.x * blockDim.x + threadIdx.x;
    long long total = (long long)depth * dRows * dCols;
    if (idx >= total) return;
    int d  = (int)(idx / ((long long)dRows * dCols));
    int rc = (int)(idx % ((long long)dRows * dCols));
    int r = rc / dCols, c = rc % dCols;
    float v = 0.f;
    if (r < sRows && c < sCols)
        v = src[(long long)d * sStride + (long long)r * sCols + c];
    dst[(long long)d * dStride + rc] = v;
}

// ---------------------------------------------------------------------------
// LayerNorm over last axis (427 valid), one row per 128-thread block,
// output zero-padded to 448 cols; rows >= 427 written as zero.
// ---------------------------------------------------------------------------
__global__ void ln_rows_kernel(const float* __restrict__ src,
                               const float* __restrict__ g,
                               const float* __restrict__ bb,
                               float* __restrict__ dst) {
    int blk = blockIdx.x;
    int b = blk / PD, r = blk % PD;
    const float* row  = src + ((long long)b * PD + r) * PD;
    float*       orow = dst + ((long long)b * PD + r) * PD;
    int tid = threadIdx.x;
    __shared__ float red[128];
    if (r >= DIMV) {
        for (int c = tid; c < PD; c += 128) orow[c] = 0.f;
        return;
    }
    float v[4]; float s = 0.f;
    #pragma unroll
    for (int i = 0; i < 4; ++i) {
        int c = tid + i * 128;
        v[i] = (c < DIMV) ? row[c] : 0.f;
        s += v[i];
    }
    red[tid] = s; __syncthreads();
    for (int off = 64; off > 0; off >>= 1) {
        if (tid < off) red[tid] += red[tid + off];
        __syncthreads();
    }
    float mean = red[0] * (1.0f / DIMV);
    __syncthreads();
    float sv = 0.f;
    #pragma unroll
    for (int i = 0; i < 4; ++i) {
        int c = tid + i * 128;
        float d = (c < DIMV) ? (v[i] - mean) : 0.f;
        sv += d * d;
    }
    red[tid] = sv; __syncthreads();
    for (int off = 64; off > 0; off >>= 1) {
        if (tid < off) red[tid] += red[tid + off];
        __syncthreads();
    }
    float rstd = rsqrtf(red[0] * (1.0f / DIMV) + 1e-5f);
    #pragma unroll
    for (int i = 0; i < 4; ++i) {
        int c = tid + i * 128;
        if (c < PD) orow[c] = (c < DIMV) ? ((v[i] - mean) * rstd * g[c] + bb[c]) : 0.f;
    }
}

// ---------------------------------------------------------------------------
// Final LN + zero diagonal + abs, writing compact [B,427,427] output.
// ---------------------------------------------------------------------------
__global__ void ln_final_kernel(const float* __restrict__ src,
                                const float* __restrict__ g,
                                const float* __restrict__ bb,
                                float* __restrict__ out) {
    int blk = blockIdx.x;
    int b = blk / DIMV, r = blk % DIMV;
    const float* row = src + ((long long)b * PD + r) * PD;
    int tid = threadIdx.x;
    __shared__ float red[128];
    float v[4]; float s = 0.f;
    #pragma unroll
    for (int i = 0; i < 4; ++i) {
        int c = tid + i * 128;
        v[i] = (c < DIMV) ? row[c] : 0.f;
        s += v[i];
    }
    red[tid] = s; __syncthreads();
    for (int off = 64; off > 0; off >>= 1) {
        if (tid < off) red[tid] += red[tid + off];
        __syncthreads();
    }
    float mean = red[0] * (1.0f / DIMV);
    __syncthreads();
    float sv = 0.f;
    #pragma unroll
    for (int i = 0; i < 4; ++i) {
        int c = tid + i * 128;
        float d = (c < DIMV) ? (v[i] - mean) : 0.f;
        sv += d * d;
    }
    red[tid] = sv; __syncthreads();
    for (int off = 64; off > 0; off >>= 1) {
        if (tid < off) red[tid] += red[tid + off];
        __syncthreads();
    }
    float rstd = rsqrtf(red[0] * (1.0f / DIMV) + 1e-5f);
    #pragma unroll
    for (int i = 0; i < 4; ++i) {
        int c = tid + i * 128;
        if (c < DIMV) {
            float val = (v[i] - mean) * rstd * g[c] + bb[c];
            out[((long long)b * DIMV + r) * DIMV + c] = (c == r) ? 0.f : fabsf(val);
        }
    }
}

// ---------------------------------------------------------------------------
// f32 WMMA GEMM, fully unmasked hot loop (A and B both zero-padded):
//   acc[m,n] = sum_k A[m,k] * B[k,n]
//   TRANSA: A[m,k] = Ab[k*LDA+m]  (else Ab[m*LDA+k])
// LDA/LDB/LDC are template constants -> all intra-trip load offsets are
// compile-time immediates; only 4 pointer bumps per 32-K trip.
// KP must be a multiple of 32. Wave tile 32x32; WG (256 thr) tile 64M x 128N.
// Epilogue MODE:
//   0: C[m*LDC+n]  = gelu(acc + bias[n])   (full padded store)
//   1: C[n*LDC+m] += acc + bias[n]         (transposed residual add, masked)
//   2: C[m*LDC+n] += acc + bias[n]         (residual add, masked)
//   3: C[m*LDC+n]  = acc + bias[n]         (store, masked)
// ---------------------------------------------------------------------------
template <int TRANSA, int MODE, int LDA, int LDB, int LDC>
__global__ __launch_bounds__(256) void gemm_wmma_kernel(
    const float* __restrict__ A, long long strideA,
    const float* __restrict__ Bm,
    const float* __restrict__ bias,
    float* __restrict__ C, long long strideC,
    int KP, int validM, int validN)
{
    const int lane = threadIdx.x & 31;
    const int wave = threadIdx.x >> 5;
    const int l    = lane & 15;
    const int hi   = lane >> 4;
    const int wm   = wave >> 2;   // 0..1
    const int wn   = wave & 3;    // 0..3
    const int m_base = blockIdx.y * 64 + wm * 32;
    const int n_base = blockIdx.x * 128 + wn * 32;

    const float* __restrict__ Ab = A + (long long)blockIdx.z * strideA;
    float* __restrict__       Cb = C + (long long)blockIdx.z * strideC;

    v8f acc[2][2] = {};

    const int k0 = hi << 1;          // this half-wave's K phase: 0 or 2
    const float* pa0;
    const float* pa1;
    if (TRANSA) {
        pa0 = Ab + k0 * LDA + (m_base + l);
        pa1 = Ab + k0 * LDA + (m_base + 16 + l);
    } else {
        pa0 = Ab + (m_base + l) * LDA + k0;
        pa1 = Ab + (m_base + 16 + l) * LDA + k0;
    }
    const float* pb0 = Bm + k0 * LDB + (n_base + l);
    const float* pb1 = Bm + k0 * LDB + (n_base + 16 + l);

    const int steps = KP >> 5;       // 32 K-values per trip
    for (int kt = 0; kt < steps; ++kt) {
        #pragma unroll
        for (int u = 0; u < 8; ++u) {
            v2f a[2], b[2];
            if (TRANSA) {
                a[0].x = pa0[(u * 4) * LDA]; a[0].y = pa0[(u * 4 + 1) * LDA];
                a[1].x = pa1[(u * 4) * LDA]; a[1].y = pa1[(u * 4 + 1) * LDA];
            } else {
                // contiguous, 8B-aligned pairs -> b64 loads with imm offsets
                float2 av0 = *reinterpret_cast<const float2*>(pa0 + u * 4);
                float2 av1 = *reinterpret_cast<const float2*>(pa1 + u * 4);
                a[0].x = av0.x; a[0].y = av0.y;
                a[1].x = av1.x; a[1].y = av1.y;
            }
            b[0].x = pb0[(u * 4) * LDB]; b[0].y = pb0[(u * 4 + 1) * LDB];
            b[1].x = pb1[(u * 4) * LDB]; b[1].y = pb1[(u * 4 + 1) * LDB];

            #pragma unroll
            for (int i = 0; i < 2; ++i)
                #pragma unroll
                for (int j = 0; j < 2; ++j)
                    acc[i][j] = __builtin_amdgcn_wmma_f32_16x16x4_f32(
                        false, (i == 0) ? a[0] : a[1], false, (j == 0) ? b[0] : b[1],
                        (short)0, acc[i][j], false, false);
        }
        pa0 += TRANSA ? 32 * LDA : 32;
        pa1 += TRANSA ? 32 * LDA : 32;
        pb0 += 32 * LDB;
        pb1 += 32 * LDB;
    }

    #pragma unroll
    for (int i = 0; i < 2; ++i) {
        #pragma unroll
        for (int j = 0; j < 2; ++j) {
            int n = n_base + j * 16 + l;
            #pragma unroll
            for (int r = 0; r < 8; ++r) {
                int m = m_base + i * 16 + hi * 8 + r;
                float vv = acc[i][j][r];
                if (MODE == 0) {
                    Cb[m * LDC + n] = gelu_exact(vv + bias[n]);
                } else if (MODE == 1) {
                    if (m < validM && n < validN) Cb[n * LDC + m] += vv + bias[n];
                } else if (MODE == 2) {
                    if (m < validM && n < validN) Cb[m * LDC + n] += vv + bias[n];
                } else {
                    if (m < validM && n < validN) Cb[m * LDC + n] = vv + bias[n];
                }
            }
        }
    }
}

// ---------------------------------------------------------------------------
extern "C" void kernel_launch(void* const* d_in, const int* in_sizes, int n_in,
                              void* d_out, int out_size, void* d_ws, size_t ws_size,
                              hipStream_t stream) {
    const float* x     = (const float*)d_in[0];
    const float* ln1_g = (const float*)d_in[1];
    const float* ln1_b = (const float*)d_in[2];
    const float* rw1   = (const float*)d_in[3];
    const float* rb1   = (const float*)d_in[4];
    const float* rw2   = (const float*)d_in[5];
    const float* rb2   = (const float*)d_in[6];
    const float* ln2_g = (const float*)d_in[7];
    const float* ln2_b = (const float*)d_in[8];
    const float* cw1   = (const float*)d_in[9];
    const float* cb1   = (const float*)d_in[10];
    const float* cw2   = (const float*)d_in[11];
    const float* cb2   = (const float*)d_in[12];
    const float* lnf_g = (const float*)d_in[13];
    const float* lnf_b = (const float*)d_in[14];
    const float* lw    = (const float*)d_in[15];
    const float* lb    = (const float*)d_in[16];
    float* out = (float*)d_out;
    (void)in_sizes; (void)n_in; (void)out_size;

    // ---- workspace layout (floats) ----
    const long long NPAD  = (long long)BATCH * PD * PD;        // 12,845,056
    const long long NT1G  = (long long)BATCH * PD * TOKC;      // 58,720,256
    const long long NW1P  = (long long)DEPTH_N * KPAD * TOKC;  //  7,340,032
    const long long NW2P  = (long long)DEPTH_N * TOKC * NPADW; //  8,388,608
    const long long NLWP  = (long long)KPAD * NPADW;           //    229,376
    const long long TOTAL = 2 * NPAD + NT1G + 2 * NW1P + 2 * NW2P + NLWP;
    if (ws_size < (size_t)TOTAL * sizeof(float)) return;

    float* hpad = (float*)d_ws;
    float* lnb  = hpad + NPAD;
    float* t1g  = lnb + NPAD;
    float* fbuf = t1g;              // reused after the depth loop
    float* rw1p = t1g + NT1G;
    float* rw2p = rw1p + NW1P;
    float* cw1p = rw2p + NW2P;
    float* cw2p = cw1p + NW1P;
    float* lwp  = cw2p + NW2P;

    // ---- one-time pads (graph-safe, deterministic) ----
    {
        long long t;
        t = (long long)DEPTH_N * KPAD * TOKC;
        pad_w_kernel<<<(int)((t + 255) / 256), 256, 0, stream>>>(
            rw1, (long long)DIMV * TOKC, DIMV, TOKC, rw1p, (long long)KPAD * TOKC, KPAD, TOKC, DEPTH_N);
        pad_w_kernel<<<(int)((t + 255) / 256), 256, 0, stream>>>(
            cw1, (long long)DIMV * TOKC, DIMV, TOKC, cw1p, (long long)KPAD * TOKC, KPAD, TOKC, DEPTH_N);
        t = (long long)DEPTH_N * TOKC * NPADW;
        pad_w_kernel<<<(int)((t + 255) / 256), 256, 0, stream>>>(
            rw2, (long long)TOKC * DIMV, TOKC, DIMV, rw2p, (long long)TOKC * NPADW, TOKC, NPADW, DEPTH_N);
        pad_w_kernel<<<(int)((t + 255) / 256), 256, 0, stream>>>(
            cw2, (long long)TOKC * DIMV, TOKC, DIMV, cw2p, (long long)TOKC * NPADW, TOKC, NPADW, DEPTH_N);
        t = (long long)KPAD * NPADW;
        pad_w_kernel<<<(int)((t + 255) / 256), 256, 0, stream>>>(
            lw, (long long)DIMV * DIMV, DIMV, DIMV, lwp, (long long)KPAD * NPADW, KPAD, NPADW, 1);
    }

    // mean over C into padded h
    mean_c_kernel<<<(int)((NPAD + 255) / 256), 256, 0, stream>>>(x, hpad);

    dim3 gBig(TOKC / 128, PD / 64, BATCH);    // (16,7,64)
    dim3 gSmall(NPADW / 128, PD / 64, BATCH); // (4,7,64)

    for (int d = 0; d < DEPTH_N; ++d) {
        // ---- token mixer ----
        ln_rows_kernel<<<BATCH * PD, 128, 0, stream>>>(
            hpad, ln1_g + d * DIMV, ln1_b + d * DIMV, lnb);
        // T1g[b,w,t] = gelu( sum_h LN1[b,h,w] * rw1[h,t] + rb1[t] )   (transA)
        gemm_wmma_kernel<1, 0, PD, TOKC, TOKC><<<gBig, 256, 0, stream>>>(
            lnb, (long long)PD * PD,
            rw1p + (long long)d * KPAD * TOKC,
            rb1 + (long long)d * TOKC,
            t1g, (long long)PD * TOKC,
            KPAD, PD, TOKC);
        // h[b,h',w] += sum_t T1g[b,w,t] * rw2[t,h'] + rb2[h']   (transposed add)
        gemm_wmma_kernel<0, 1, TOKC, NPADW, PD><<<gSmall, 256, 0, stream>>>(
            t1g, (long long)PD * TOKC,
            rw2p + (long long)d * TOKC * NPADW,
            rb2 + (long long)d * DIMV,
            hpad, (long long)PD * PD,
            TOKC, DIMV, DIMV);
        // ---- channel mixer ----
        ln_rows_kernel<<<BATCH * PD, 128, 0, stream>>>(
            hpad, ln2_g + d * DIMV, ln2_b + d * DIMV, lnb);
        // T1g[b,i,t] = gelu( sum_j LN2[b,i,j] * cw1[j,t] + cb1[t] )
        gemm_wmma_kernel<0, 0, PD, TOKC, TOKC><<<gBig, 256, 0, stream>>>(
            lnb, (long long)PD * PD,
            cw1p + (long long)d * KPAD * TOKC,
            cb1 + (long long)d * TOKC,
            t1g, (long long)PD * TOKC,
            KPAD, PD, TOKC);
        // h[b,i,j'] += sum_t T1g[b,i,t] * cw2[t,j'] + cb2[j']
        gemm_wmma_kernel<0, 2, TOKC, NPADW, PD><<<gSmall, 256, 0, stream>>>(
            t1g, (long long)PD * TOKC,
            cw2p + (long long)d * TOKC * NPADW,
            cb2 + (long long)d * DIMV,
            hpad, (long long)PD * PD,
            TOKC, DIMV, DIMV);
    }

    // ---- final: LN -> linear(+bias) -> LN -> mask diag -> abs ----
    ln_rows_kernel<<<BATCH * PD, 128, 0, stream>>>(hpad, lnf_g, lnf_b, lnb);
    gemm_wmma_kernel<0, 3, PD, NPADW, PD><<<gSmall, 256, 0, stream>>>(
        lnb, (long long)PD * PD,
        lwp, lb,
        fbuf, (long long)PD * PD,
        KPAD, DIMV, DIMV);
    ln_final_kernel<<<BATCH * DIMV, 128, 0, stream>>>(fbuf, lnf_g, lnf_b, out);
}